// ConcatHandshaking_30605936951494
// MI455X (gfx1250) — compile-verified
//
#include <hip/hip_runtime.h>
#include <math.h>

typedef float v2f __attribute__((ext_vector_type(2)));
typedef float v4f __attribute__((ext_vector_type(4)));
typedef float v8f __attribute__((ext_vector_type(8)));

static constexpr int kH     = 768;    // hidden dim
static constexpr int kS     = 256;    // sequence length
static constexpr int kB     = 4;      // batch
static constexpr int kN2    = 1536;   // 2*H (P row stride: [P1 | P2])
static constexpr int kPairs = 32896;  // S*(S+1)/2
static constexpr int kTN    = 96;     // N tiles (1536/16)

// -----------------------------------------------------------------------------
// Kernel 1: P[m, n] for m in [0,1024), n in [0,1536)
//   n <  768 : sum_h X[m,h] * W[n, h]        + bias[n]   (= p1 + fc_b)
//   n >= 768 : sum_h X[m,h] * W[n-768, h+768]            (= p2)
// One wave32 computes one 16x16 f32 tile via V_WMMA_F32_16X16X4_F32.
// Grid covers tiles exactly -> EXEC all-ones at every WMMA.
// -----------------------------------------------------------------------------
__global__ __launch_bounds__(256) void gemm_bias_wmma(
    const float* __restrict__ X, const float* __restrict__ W,
    const float* __restrict__ bias, float* __restrict__ P)
{
    const int lane = threadIdx.x & 31;
    const int wave = threadIdx.x >> 5;
    const int tile = blockIdx.x * 8 + wave;   // 768 blocks * 8 waves = 6144 tiles
    const int tN = tile % kTN;
    const int tM = tile / kTN;
    const int m0 = tM * 16;
    const int n0 = tN * 16;

    const int halfSel = lane >> 4;   // 0: lanes 0-15 (K=0,1), 1: lanes 16-31 (K=2,3)
    const int l15     = lane & 15;

    // A-frag (16x4): lane reads X[m0+l15, k + 2*halfSel .. +1]  (8B aligned)
    const float* aPtr = X + (size_t)(m0 + l15) * kH + halfSel * 2;

    // B-frag (4x16): B[k, n] = Wcat[n, k]; column n fixed per lane.
    const int n = n0 + l15;                 // n<768 uniform per wave (tiles don't straddle)
    const float* bPtr = (n < kH) ? (W + (size_t)n * kN2 + halfSel * 2)
                                 : (W + (size_t)(n - kH) * kN2 + kH + halfSel * 2);

    v8f acc = {0.f, 0.f, 0.f, 0.f, 0.f, 0.f, 0.f, 0.f};
    for (int k = 0; k < kH; k += 4) {
        v2f a = *(const v2f*)(aPtr + k);
        v2f b = *(const v2f*)(bPtr + k);
        // D = A(16x4) x B(4x16) + C ; 8 args: neg_a, A, neg_b, B, c_mod, C, reuse_a, reuse_b
        acc = __builtin_amdgcn_wmma_f32_16x16x4_f32(false, a, false, b,
                                                    (short)0, acc, false, false);
    }

    // Fold bias into the P1 half (bias is per output column o = n).
    if (n < kH) {
        const float bn = bias[n];
        #pragma unroll
        for (int r = 0; r < 8; ++r) acc[r] += bn;
    }

    // D layout: lane holds column l15; VGPR r holds row r (lanes 0-15) / r+8 (16-31).
    float* dst = P + (size_t)(m0 + halfSel * 8) * kN2 + n0 + l15;
    #pragma unroll
    for (int r = 0; r < 8; ++r) dst[(size_t)r * kN2] = acc[r];
}

// -----------------------------------------------------------------------------
// Kernel 2: out[b, p, :] = tanh(P1[b, i_p, :] + P2[b, j_p, :])   (bias already in P1)
// One 192-thread block per (b, pair); each thread handles one float4 (768 = 192*4).
// Output is write-once (404 MB) -> non-temporal stores. P rows hit L2 (6 MB total).
// -----------------------------------------------------------------------------
__device__ __forceinline__ int triu_row_start(int i) {
    return i * kS - (i * (i - 1)) / 2;   // # pairs before row i
}

__global__ __launch_bounds__(192) void pair_tanh(
    const float* __restrict__ P, float* __restrict__ out)
{
    const int p = blockIdx.x;   // pair index, row-major upper triangle (i<=j)
    const int b = blockIdx.y;

    // Invert p -> (i, j): closed form + integer fixup.
    const double t = 2.0 * kS + 1.0;
    int i = (int)((t - sqrt(t * t - 8.0 * (double)p)) * 0.5);
    if (i < 0) i = 0;
    if (i > kS - 1) i = kS - 1;
    while (triu_row_start(i + 1) <= p) ++i;
    while (triu_row_start(i) > p) --i;
    const int j = i + (p - triu_row_start(i));

    const v4f* p1r = (const v4f*)(P + ((size_t)b * kS + i) * kN2);        // P1 row
    const v4f* p2r = (const v4f*)(P + ((size_t)b * kS + j) * kN2 + kH);   // P2 row
    v4f* orow = (v4f*)(out + ((size_t)b * kPairs + p) * (size_t)kH);

    const int o = threadIdx.x;          // 0..191 float4 chunks
    const v4f v1 = p1r[o];
    const v4f v2 = p2r[o];
    v4f r;
    r.x = tanhf(v1.x + v2.x);
    r.y = tanhf(v1.y + v2.y);
    r.z = tanhf(v1.z + v2.z);
    r.w = tanhf(v1.w + v2.w);
    __builtin_nontemporal_store(r, orow + o);
}

// -----------------------------------------------------------------------------
extern "C" void kernel_launch(void* const* d_in, const int* in_sizes, int n_in,
                              void* d_out, int out_size, void* d_ws, size_t ws_size,
                              hipStream_t stream) {
    const float* X    = (const float*)d_in[0];   // (4, 256, 768)
    const float* W    = (const float*)d_in[1];   // (768, 1536)
    const float* bias = (const float*)d_in[2];   // (768,)
    float* out = (float*)d_out;                  // (4, 32896, 768) f32
    float* P   = (float*)d_ws;                   // 1024 x 1536 f32 = 6 MB scratch

    // 6144 tiles / 8 waves per 256-thread block
    gemm_bias_wmma<<<768, 256, 0, stream>>>(X, W, bias, P);

    dim3 grid2(kPairs, kB);
    pair_tanh<<<grid2, 192, 0, stream>>>(P, out);
}